// SparseMiddleLayer_69587060129988
// MI455X (gfx1250) — compile-verified
//
#include <hip/hip_runtime.h>
#include <hip/hip_bf16.h>

// ---------------------------------------------------------------------------
// CDNA5 (gfx1250) sparse-middle-encoder, round 2.
//  * All GEMMs on V_WMMA_F32_16X16X4_F32 (fp32-exact vs reference).
//  * Activations kept NDHWC so each A fragment is ONE global_load_b64.
//  * Weights repacked once per launch to [cot][tap][c4][16][4] so each B
//    fragment is ONE global_load_b64 with an immediate offset.
//  * 2x2 register tile per wave: 4 b64 loads -> 4 WMMAs (1 vmem : 1 wmma).
//  * No divergent loads anywhere in the hot loops: safe-address + v_cndmask.
// ---------------------------------------------------------------------------

typedef __attribute__((ext_vector_type(2))) float v2f;
typedef __attribute__((ext_vector_type(8))) float v8f;

#define BN_EPS      0.001f
#define LEAKY_SLOPE 0.1f

__device__ __forceinline__ float leaky(float x) {
    return x >= 0.f ? x : LEAKY_SLOPE * x;
}

constexpr int BATCH = 2;
constexpr int D0 = 41, H0 = 128, W0 = 112;   // sparse shape

// ============================================================================
// Stage A: voxel grid / neighbor table
// ============================================================================
__global__ void scatter_grid_kernel(const int* __restrict__ coors,
                                    int* __restrict__ grid, int N) {
    int n = blockIdx.x * blockDim.x + threadIdx.x;
    if (n >= N) return;
    int b = coors[4 * n + 0], z = coors[4 * n + 1];
    int y = coors[4 * n + 2], x = coors[4 * n + 3];
    grid[((b * D0 + z) * H0 + y) * W0 + x] = n;
}

__global__ void build_nbr_kernel(const int* __restrict__ coors,
                                 const int* __restrict__ grid,
                                 int* __restrict__ nbr, int N) {
    int t = blockIdx.x * blockDim.x + threadIdx.x;
    if (t >= N * 27) return;
    int n = t / 27, k = t % 27;
    int dz = k / 9 - 1, dy = (k / 3) % 3 - 1, dx = k % 3 - 1;
    int b = coors[4 * n + 0];
    int z = coors[4 * n + 1] + dz;
    int y = coors[4 * n + 2] + dy;
    int x = coors[4 * n + 3] + dx;
    int idx = -1;
    if (z >= 0 && z < D0 && y >= 0 && y < H0 && x >= 0 && x < W0)
        idx = grid[((b * D0 + z) * H0 + y) * W0 + x];
    nbr[t] = idx;
}

// ============================================================================
// Weight repack kernels (run once per launch; tiny).
// Dense conv OIDHW (Cout,Cin,KD,KH,KW) ->
//   rw[(((cot*KDHW+tap)*C4 + c4)*16 + col)*4 + kv] = w[((cot*16+col)*Cin + c4*4+kv)*KDHW + tap]
// so a lane's B fragment (K = half*2 + v, N = col) is one contiguous b64.
// ============================================================================
__global__ void repack_conv_w_kernel(const float* __restrict__ w,
                                     float* __restrict__ rw,
                                     int Cout, int Cin, int KDHW) {
    int t = blockIdx.x * blockDim.x + threadIdx.x;
    int tot = Cout * Cin * KDHW;
    if (t >= tot) return;
    int C4 = Cin >> 2;
    int kv = t & 3;
    int col = (t >> 2) & 15;
    int u = t >> 6;
    int c4 = u % C4; u /= C4;
    int tap = u % KDHW; int cot = u / KDHW;
    rw[t] = w[((cot * 16 + col) * Cin + c4 * 4 + kv) * KDHW + tap];
}

// Subm weights (27, CIN, 16) -> rs[((k*C4+c4)*16+col)*4+kv] = w[(k*CIN+c4*4+kv)*16+col]
__global__ void repack_subm_w_kernel(const float* __restrict__ w,
                                     float* __restrict__ rs, int Cin) {
    int t = blockIdx.x * blockDim.x + threadIdx.x;
    int tot = 27 * Cin * 16;
    if (t >= tot) return;
    int C4 = Cin >> 2;
    int kv = t & 3;
    int col = (t >> 2) & 15;
    int u = t >> 6;
    int c4 = u % C4; int k = u / C4;
    rs[t] = w[(k * Cin + c4 * 4 + kv) * 16 + col];
}

// ============================================================================
// Submanifold sparse conv: gather-GEMM, 2 M-subtiles (32 voxels) x 16 chans
// per wave sharing the B fragment. feats rows are contiguous (n, CIN).
// ============================================================================
template <int CIN>
__global__ void subm_wmma_kernel(const float* __restrict__ feats,
                                 const int* __restrict__ nbr,
                                 const float* __restrict__ rs,  // repacked
                                 const float* __restrict__ bng,
                                 const float* __restrict__ bnb,
                                 const float* __restrict__ bnm,
                                 const float* __restrict__ bnv,
                                 float* __restrict__ out, int N) {
    constexpr int C4 = CIN / 4;
    const int lane = threadIdx.x;
    const int col  = lane & 15;
    const int half = lane >> 4;
    const int tile = blockIdx.x;
    const int r0 = tile * 32 + col;
    const int r1 = r0 + 16;
    const bool pv0 = (r0 < N), pv1 = (r1 < N);
    const int q0 = pv0 ? r0 : 0, q1 = pv1 ? r1 : 0;
    const v2f zz = {0.f, 0.f};

    v8f acc0 = {}, acc1 = {};
    for (int k = 0; k < 27; ++k) {
        // unconditional gather-table loads from safe indices
        int i0 = nbr[q0 * 27 + k];
        int i1 = nbr[q1 * 27 + k];
        bool ok0 = pv0 && (i0 >= 0);
        bool ok1 = pv1 && (i1 >= 0);
        const float* ap0 = feats + (size_t)(ok0 ? i0 : 0) * CIN + half * 2;
        const float* ap1 = feats + (size_t)(ok1 ? i1 : 0) * CIN + half * 2;
        const float* bp  = rs + (size_t)k * C4 * 64 + col * 4 + half * 2;
#pragma unroll 4
        for (int c4 = 0; c4 < C4; ++c4) {
            v2f a0 = *(const v2f*)(ap0 + c4 * 4);
            v2f a1 = *(const v2f*)(ap1 + c4 * 4);
            v2f bf = *(const v2f*)(bp + c4 * 64);
            a0 = ok0 ? a0 : zz;
            a1 = ok1 ? a1 : zz;
            acc0 = __builtin_amdgcn_wmma_f32_16x16x4_f32(
                false, a0, false, bf, (short)0, acc0, false, false);
            acc1 = __builtin_amdgcn_wmma_f32_16x16x4_f32(
                false, a1, false, bf, (short)0, acc1, false, false);
        }
    }
    float g = bng[col], bb = bnb[col], mu = bnm[col], var = bnv[col];
    float scale = g * rsqrtf(var + BN_EPS);
#pragma unroll
    for (int j = 0; j < 8; ++j) {
        int ra = tile * 32 + j + 8 * half;
        int rb = ra + 16;
        if (ra < N) out[ra * 16 + col] = leaky((acc0[j] - mu) * scale + bb);
        if (rb < N) out[rb * 16 + col] = leaky((acc1[j] - mu) * scale + bb);
    }
}

// ============================================================================
// Scatter voxels into dense NDHWC tensor (+ mask); rows are 16 contiguous ch.
// ============================================================================
__global__ void scatter_dense_kernel(const float* __restrict__ f,
                                     const int* __restrict__ coors,
                                     float* __restrict__ xdense,  // NDHWC, C=16
                                     float* __restrict__ mask, int N) {
    int n = blockIdx.x * blockDim.x + threadIdx.x;
    if (n >= N) return;
    int b = coors[4 * n + 0], z = coors[4 * n + 1];
    int y = coors[4 * n + 2], x = coors[4 * n + 3];
    int sp = ((b * D0 + z) * H0 + y) * W0 + x;
    const float4* src = (const float4*)(f + (size_t)n * 16);
    float4* dst = (float4*)(xdense + (size_t)sp * 16);
    dst[0] = src[0]; dst[1] = src[1]; dst[2] = src[2]; dst[3] = src[3];
    mask[sp] = 1.f;
}

// ============================================================================
// Mask downsample: out = (conv(mask, ones, stride, pad) > 0)
// ============================================================================
__global__ void mask_down_kernel(const float* __restrict__ min_,
                                 float* __restrict__ mout,
                                 int B, int Di, int Hi, int Wi,
                                 int Do, int Ho, int Wo,
                                 int KD, int KH, int KW,
                                 int sd, int sh, int sw,
                                 int pd, int ph, int pw) {
    int t = blockIdx.x * blockDim.x + threadIdx.x;
    int tot = B * Do * Ho * Wo;
    if (t >= tot) return;
    int x = t % Wo, tt = t / Wo;
    int y = tt % Ho; tt /= Ho;
    int z = tt % Do; int b = tt / Do;
    float any = 0.f;
    for (int kd = 0; kd < KD; ++kd) {
        int zi = z * sd - pd + kd;
        if (zi < 0 || zi >= Di) continue;
        for (int kh = 0; kh < KH; ++kh) {
            int yi = y * sh - ph + kh;
            if (yi < 0 || yi >= Hi) continue;
            for (int kw = 0; kw < KW; ++kw) {
                int xi = x * sw - pw + kw;
                if (xi < 0 || xi >= Wi) continue;
                any += min_[((b * Di + zi) * Hi + yi) * Wi + xi];
            }
        }
    }
    mout[t] = (any > 0.f) ? 1.f : 0.f;
}

// ============================================================================
// Dense 3D conv, implicit GEMM, NDHWC activations, repacked weights.
// One wave computes a 32(M) x 32(N) tile as 2x2 WMMA subtiles.
// grid = (ceil(M/32), Cout/32). Epilogue: BN + leaky + mask (mask idx == r).
// nchw_out=1 only for the final layer (writes d_out in reference layout).
// ============================================================================
template <int CIN>
__global__ void conv3d_wmma_kernel(const float* __restrict__ x,    // NDHWC
                                   const float* __restrict__ rw,   // repacked
                                   const float* __restrict__ bng,
                                   const float* __restrict__ bnb,
                                   const float* __restrict__ bnm,
                                   const float* __restrict__ bnv,
                                   const float* __restrict__ maskOut,
                                   float* __restrict__ out,
                                   int B, int Di, int Hi, int Wi,
                                   int Cout, int Do, int Ho, int Wo,
                                   int KD, int KH, int KW,
                                   int sd, int sh, int sw,
                                   int pd, int ph, int pw,
                                   int nchw_out) {
    constexpr int C4 = CIN / 4;
    const int lane = threadIdx.x;
    const int col  = lane & 15;
    const int half = lane >> 4;
    const int M    = B * Do * Ho * Wo;
    const int tile = blockIdx.x;
    const int KDHW = KD * KH * KW;
    const int HiWi = Hi * Wi;
    const v2f zz = {0.f, 0.f};

    // two M-subtiles: decompose this lane's two A-row positions once
    int r0 = tile * 32 + col, r1 = r0 + 16;
    bool pv0 = (r0 < M), pv1 = (r1 < M);
    int q0 = pv0 ? r0 : 0, q1 = pv1 ? r1 : 0;
    int xo0 = q0 % Wo, u0 = q0 / Wo, yo0 = u0 % Ho, v0 = u0 / Ho;
    int zo0 = v0 % Do, b0 = v0 / Do;
    int xo1 = q1 % Wo, u1 = q1 / Wo, yo1 = u1 % Ho, v1 = u1 / Ho;
    int zo1 = v1 % Do, b1 = v1 / Do;
    int zib0 = zo0 * sd - pd, yib0 = yo0 * sh - ph, xib0 = xo0 * sw - pw;
    int zib1 = zo1 * sd - pd, yib1 = yo1 * sh - ph, xib1 = xo1 * sw - pw;
    const float* xb0 = x + (size_t)b0 * Di * HiWi * CIN + half * 2;
    const float* xb1 = x + (size_t)b1 * Di * HiWi * CIN + half * 2;

    // two N-subtiles
    const int cot0 = blockIdx.y * 2, cot1 = cot0 + 1;
    const int co0 = blockIdx.y * 32 + col, co1 = co0 + 16;

    v8f d00 = {}, d01 = {}, d10 = {}, d11 = {};
    for (int kd = 0; kd < KD; ++kd) {
        int zi0 = zib0 + kd, zi1 = zib1 + kd;
        bool okz0 = (zi0 >= 0 && zi0 < Di), okz1 = (zi1 >= 0 && zi1 < Di);
        for (int kh = 0; kh < KH; ++kh) {
            int yi0 = yib0 + kh, yi1 = yib1 + kh;
            bool oky0 = (yi0 >= 0 && yi0 < Hi), oky1 = (yi1 >= 0 && yi1 < Hi);
            for (int kw = 0; kw < KW; ++kw) {
                int xi0 = xib0 + kw, xi1 = xib1 + kw;
                bool ok0 = pv0 && okz0 && oky0 && (xi0 >= 0 && xi0 < Wi);
                bool ok1 = pv1 && okz1 && oky1 && (xi1 >= 0 && xi1 < Wi);
                int s0 = ok0 ? (zi0 * HiWi + yi0 * Wi + xi0) : 0;
                int s1 = ok1 ? (zi1 * HiWi + yi1 * Wi + xi1) : 0;
                const float* ap0 = xb0 + (size_t)s0 * CIN;
                const float* ap1 = xb1 + (size_t)s1 * CIN;
                int tap = (kd * KH + kh) * KW + kw;
                const float* bp0 = rw + (size_t)(cot0 * KDHW + tap) * C4 * 64
                                      + col * 4 + half * 2;
                const float* bp1 = rw + (size_t)(cot1 * KDHW + tap) * C4 * 64
                                      + col * 4 + half * 2;
#pragma unroll 4
                for (int c4 = 0; c4 < C4; ++c4) {
                    v2f a0 = *(const v2f*)(ap0 + c4 * 4);
                    v2f a1 = *(const v2f*)(ap1 + c4 * 4);
                    v2f w0 = *(const v2f*)(bp0 + c4 * 64);
                    v2f w1 = *(const v2f*)(bp1 + c4 * 64);
                    a0 = ok0 ? a0 : zz;
                    a1 = ok1 ? a1 : zz;
                    d00 = __builtin_amdgcn_wmma_f32_16x16x4_f32(
                        false, a0, false, w0, (short)0, d00, false, false);
                    d01 = __builtin_amdgcn_wmma_f32_16x16x4_f32(
                        false, a0, false, w1, (short)0, d01, false, false);
                    d10 = __builtin_amdgcn_wmma_f32_16x16x4_f32(
                        false, a1, false, w0, (short)0, d10, false, false);
                    d11 = __builtin_amdgcn_wmma_f32_16x16x4_f32(
                        false, a1, false, w1, (short)0, d11, false, false);
                }
            }
        }
    }

    // epilogue: BN + leaky + mask; NDHWC store (coalesced) or NCDHW for final
    float g0 = bng[co0], be0 = bnb[co0], mu0 = bnm[co0], va0 = bnv[co0];
    float g1 = bng[co1], be1 = bnb[co1], mu1 = bnm[co1], va1 = bnv[co1];
    float sc0 = g0 * rsqrtf(va0 + BN_EPS);
    float sc1 = g1 * rsqrtf(va1 + BN_EPS);
    const int HoWo = Ho * Wo;

    auto emit = [&](float val, int r, int co) {
        if (nchw_out) {
            int x2 = r % Wo, tt = r / Wo;
            int y2 = tt % Ho; tt /= Ho;
            int z2 = tt % Do; int b2 = tt / Do;
            out[((b2 * (size_t)Cout + co) * Do + z2) * HoWo + y2 * Wo + x2] = val;
        } else {
            out[(size_t)r * Cout + co] = val;
        }
    };
#pragma unroll
    for (int j = 0; j < 8; ++j) {
        int ra = tile * 32 + j + 8 * half;
        int rb = ra + 16;
        if (ra < M) {
            float msk = maskOut[ra];
            emit(leaky((d00[j] - mu0) * sc0 + be0) * msk, ra, co0);
            emit(leaky((d01[j] - mu1) * sc1 + be1) * msk, ra, co1);
        }
        if (rb < M) {
            float msk = maskOut[rb];
            emit(leaky((d10[j] - mu0) * sc0 + be0) * msk, rb, co0);
            emit(leaky((d11[j] - mu1) * sc1 + be1) * msk, rb, co1);
        }
    }
}

// ============================================================================
// Host orchestration
// ============================================================================
extern "C" void kernel_launch(void* const* d_in, const int* in_sizes, int n_in,
                              void* d_out, int out_size, void* d_ws, size_t ws_size,
                              hipStream_t stream) {
    const float* vf    = (const float*)d_in[0];
    const int*   coors = (const int*)d_in[1];
    const int N = in_sizes[0] / 128;   // 60000 voxels

    // params flattened in dict insertion order:
    // [3]=w0a [4]=w0b [5]=w1 [6]=ws1a [7]=ws1b [8]=w2 [9..11]=ws2a/b/c
    // [12]=w3 [13..15]=ws3a/b/c [16]=w4, then bn[i]: g,b,m,v at 17+4i..
    const float* w0a  = (const float*)d_in[3];
    const float* w0b  = (const float*)d_in[4];
    const float* w1   = (const float*)d_in[5];
    const float* ws1a = (const float*)d_in[6];
    const float* ws1b = (const float*)d_in[7];
    const float* w2   = (const float*)d_in[8];
    const float* ws2a = (const float*)d_in[9];
    const float* ws2b = (const float*)d_in[10];
    const float* ws2c = (const float*)d_in[11];
    const float* w3   = (const float*)d_in[12];
    const float* ws3a = (const float*)d_in[13];
    const float* ws3b = (const float*)d_in[14];
    const float* ws3c = (const float*)d_in[15];
    const float* w4   = (const float*)d_in[16];
    const float *bnG[14], *bnB[14], *bnM[14], *bnV[14];
    for (int i = 0; i < 14; ++i) {
        bnG[i] = (const float*)d_in[17 + 4 * i + 0];
        bnB[i] = (const float*)d_in[17 + 4 * i + 1];
        bnM[i] = (const float*)d_in[17 + 4 * i + 2];
        bnV[i] = (const float*)d_in[17 + 4 * i + 3];
    }

    constexpr int D1 = 21, H1 = 64, W1 = 56;
    constexpr int D2 = 11, H2 = 32, W2 = 28;
    constexpr int D3 = 5,  H3 = 16, W3 = 14;
    constexpr int D4 = 2,  H4 = 16, W4 = 14;

    char* wsp = (char*)d_ws;
    auto carve = [&](size_t bytes) -> void* {
        void* pp = (void*)wsp;
        wsp += (bytes + 255) & ~size_t(255);
        return pp;
    };
    int*   grid  = (int*)carve((size_t)BATCH * D0 * H0 * W0 * 4);
    int*   nbr   = (int*)carve((size_t)N * 27 * 4);
    float* f1    = (float*)carve((size_t)N * 16 * 4);
    float* f2    = (float*)carve((size_t)N * 16 * 4);
    float* x0    = (float*)carve((size_t)BATCH * 16 * D0 * H0 * W0 * 4);
    float* mask0 = (float*)carve((size_t)BATCH * D0 * H0 * W0 * 4);
    float* x1a   = (float*)carve((size_t)BATCH * 32 * D1 * H1 * W1 * 4);
    float* x1b   = (float*)carve((size_t)BATCH * 32 * D1 * H1 * W1 * 4);
    float* mask1 = (float*)carve((size_t)BATCH * D1 * H1 * W1 * 4);
    float* x2a   = (float*)carve((size_t)BATCH * 64 * D2 * H2 * W2 * 4);
    float* x2b   = (float*)carve((size_t)BATCH * 64 * D2 * H2 * W2 * 4);
    float* mask2 = (float*)carve((size_t)BATCH * D2 * H2 * W2 * 4);
    float* x3a   = (float*)carve((size_t)BATCH * 64 * D3 * H3 * W3 * 4);
    float* x3b   = (float*)carve((size_t)BATCH * 64 * D3 * H3 * W3 * 4);
    float* mask3 = (float*)carve((size_t)BATCH * D3 * H3 * W3 * 4);
    float* mask4 = (float*)carve((size_t)BATCH * D4 * H4 * W4 * 4);
    // repacked weights
    float* rs0a = (float*)carve((size_t)27 * 128 * 16 * 4);
    float* rs0b = (float*)carve((size_t)27 * 16 * 16 * 4);
    float* rw1  = (float*)carve((size_t)32 * 16 * 27 * 4);
    float* rw1a = (float*)carve((size_t)32 * 32 * 27 * 4);
    float* rw1b = (float*)carve((size_t)32 * 32 * 27 * 4);
    float* rw2  = (float*)carve((size_t)64 * 32 * 27 * 4);
    float* rw2a = (float*)carve((size_t)64 * 64 * 27 * 4);
    float* rw2b = (float*)carve((size_t)64 * 64 * 27 * 4);
    float* rw2c = (float*)carve((size_t)64 * 64 * 27 * 4);
    float* rw3  = (float*)carve((size_t)64 * 64 * 27 * 4);
    float* rw3a = (float*)carve((size_t)64 * 64 * 27 * 4);
    float* rw3b = (float*)carve((size_t)64 * 64 * 27 * 4);
    float* rw3c = (float*)carve((size_t)64 * 64 * 27 * 4);
    float* rw4  = (float*)carve((size_t)64 * 64 * 3 * 4);
    (void)ws_size;

    auto repackc = [&](const float* src, float* dst, int Cout, int Cin, int KDHW) {
        int tot = Cout * Cin * KDHW;
        repack_conv_w_kernel<<<(tot + 255) / 256, 256, 0, stream>>>(
            src, dst, Cout, Cin, KDHW);
    };
    repack_subm_w_kernel<<<(27 * 128 * 16 + 255) / 256, 256, 0, stream>>>(w0a, rs0a, 128);
    repack_subm_w_kernel<<<(27 * 16 * 16 + 255) / 256, 256, 0, stream>>>(w0b, rs0b, 16);
    repackc(w1, rw1, 32, 16, 27);
    repackc(ws1a, rw1a, 32, 32, 27);  repackc(ws1b, rw1b, 32, 32, 27);
    repackc(w2, rw2, 64, 32, 27);
    repackc(ws2a, rw2a, 64, 64, 27);  repackc(ws2b, rw2b, 64, 64, 27);
    repackc(ws2c, rw2c, 64, 64, 27);
    repackc(w3, rw3, 64, 64, 27);
    repackc(ws3a, rw3a, 64, 64, 27);  repackc(ws3b, rw3b, 64, 64, 27);
    repackc(ws3c, rw3c, 64, 64, 27);
    repackc(w4, rw4, 64, 64, 3);

    // ---- Stage A: neighbor table + sparse subm layers ----
    hipMemsetAsync(grid, 0xFF, (size_t)BATCH * D0 * H0 * W0 * 4, stream); // -1
    scatter_grid_kernel<<<(N + 255) / 256, 256, 0, stream>>>(coors, grid, N);
    build_nbr_kernel<<<(N * 27 + 255) / 256, 256, 0, stream>>>(coors, grid, nbr, N);

    int tiles2 = (N + 31) / 32;
    subm_wmma_kernel<128><<<tiles2, 32, 0, stream>>>(
        vf, nbr, rs0a, bnG[0], bnB[0], bnM[0], bnV[0], f1, N);
    subm_wmma_kernel<16><<<tiles2, 32, 0, stream>>>(
        f1, nbr, rs0b, bnG[1], bnB[1], bnM[1], bnV[1], f2, N);

    // ---- Stage B: densify (NDHWC) ----
    hipMemsetAsync(x0,    0, (size_t)BATCH * 16 * D0 * H0 * W0 * 4, stream);
    hipMemsetAsync(mask0, 0, (size_t)BATCH * D0 * H0 * W0 * 4, stream);
    scatter_dense_kernel<<<(N + 255) / 256, 256, 0, stream>>>(f2, coors, x0, mask0, N);

    // ---- Stage C: dense conv chain ----
    auto mdown = [&](const float* mi, float* mo,
                     int Di, int Hi, int Wi, int Do, int Ho, int Wo,
                     int KD, int KH, int KW, int sd, int sh, int sw,
                     int pd, int ph, int pw) {
        int tot = BATCH * Do * Ho * Wo;
        mask_down_kernel<<<(tot + 255) / 256, 256, 0, stream>>>(
            mi, mo, BATCH, Di, Hi, Wi, Do, Ho, Wo,
            KD, KH, KW, sd, sh, sw, pd, ph, pw);
    };
#define CONV(CINV, in, rw_, bni, mk, outp, Di, Hi, Wi, Cout, Do, Ho, Wo,      \
             KD, KH, KW, sd, sh, sw, pd, ph, pw, nchw)                        \
    do {                                                                      \
        int M_ = BATCH * (Do) * (Ho) * (Wo);                                  \
        dim3 g_((M_ + 31) / 32, (Cout) / 32);                                 \
        conv3d_wmma_kernel<CINV><<<g_, 32, 0, stream>>>(                      \
            in, rw_, bnG[bni], bnB[bni], bnM[bni], bnV[bni], mk, outp,        \
            BATCH, Di, Hi, Wi, Cout, Do, Ho, Wo,                              \
            KD, KH, KW, sd, sh, sw, pd, ph, pw, nchw);                        \
    } while (0)

    // down 1: 16 -> 32, s2 p1
    mdown(mask0, mask1, D0, H0, W0, D1, H1, W1, 3, 3, 3, 2, 2, 2, 1, 1, 1);
    CONV(16, x0, rw1, 2, mask1, x1a, D0, H0, W0, 32, D1, H1, W1, 3, 3, 3, 2, 2, 2, 1, 1, 1, 0);
    CONV(32, x1a, rw1a, 3, mask1, x1b, D1, H1, W1, 32, D1, H1, W1, 3, 3, 3, 1, 1, 1, 1, 1, 1, 0);
    CONV(32, x1b, rw1b, 4, mask1, x1a, D1, H1, W1, 32, D1, H1, W1, 3, 3, 3, 1, 1, 1, 1, 1, 1, 0);

    // down 2: 32 -> 64, s2 p1
    mdown(mask1, mask2, D1, H1, W1, D2, H2, W2, 3, 3, 3, 2, 2, 2, 1, 1, 1);
    CONV(32, x1a, rw2, 5, mask2, x2a, D1, H1, W1, 64, D2, H2, W2, 3, 3, 3, 2, 2, 2, 1, 1, 1, 0);
    CONV(64, x2a, rw2a, 6, mask2, x2b, D2, H2, W2, 64, D2, H2, W2, 3, 3, 3, 1, 1, 1, 1, 1, 1, 0);
    CONV(64, x2b, rw2b, 7, mask2, x2a, D2, H2, W2, 64, D2, H2, W2, 3, 3, 3, 1, 1, 1, 1, 1, 1, 0);
    CONV(64, x2a, rw2c, 8, mask2, x2b, D2, H2, W2, 64, D2, H2, W2, 3, 3, 3, 1, 1, 1, 1, 1, 1, 0);

    // down 3: 64 -> 64, s2 p(0,1,1)
    mdown(mask2, mask3, D2, H2, W2, D3, H3, W3, 3, 3, 3, 2, 2, 2, 0, 1, 1);
    CONV(64, x2b, rw3, 9, mask3, x3a, D2, H2, W2, 64, D3, H3, W3, 3, 3, 3, 2, 2, 2, 0, 1, 1, 0);
    CONV(64, x3a, rw3a, 10, mask3, x3b, D3, H3, W3, 64, D3, H3, W3, 3, 3, 3, 1, 1, 1, 1, 1, 1, 0);
    CONV(64, x3b, rw3b, 11, mask3, x3a, D3, H3, W3, 64, D3, H3, W3, 3, 3, 3, 1, 1, 1, 1, 1, 1, 0);
    CONV(64, x3a, rw3c, 12, mask3, x3b, D3, H3, W3, 64, D3, H3, W3, 3, 3, 3, 1, 1, 1, 1, 1, 1, 0);

    // down 4: k(3,1,1) s(2,1,1) p0 -> d_out in NCDHW (== reference reshape)
    mdown(mask3, mask4, D3, H3, W3, D4, H4, W4, 3, 1, 1, 2, 1, 1, 0, 0, 0);
    CONV(64, x3b, rw4, 13, mask4, (float*)d_out, D3, H3, W3, 64, D4, H4, W4,
         3, 1, 1, 2, 1, 1, 0, 0, 0, 1);
#undef CONV
    (void)out_size; (void)n_in;
}